// CRF_1735166787812
// MI455X (gfx1250) — compile-verified
//
#include <hip/hip_runtime.h>

typedef __attribute__((ext_vector_type(2))) float v2f;
typedef __attribute__((ext_vector_type(8))) float v8f;

#define B_ 128
#define T_ 512
#define E_ 256
#define K_ 32

// ---------------------------------------------------------------------------
// Kernel 1: emit[b,t,:] = emb[sentences[b,t]] @ W + b
// One wave (32 lanes) computes a 16-row x 32-col tile using V_WMMA_F32_16X16X4_F32.
// 8 waves per block. W staged in LDS in k-pair-interleaved layout:
//   Wp[p][n][q] = W[2p+q][n]   (p = k/2, q = k&1)
// so each lane's B operand (W[k0][n], W[k0+1][n]) is one contiguous ds_load_b64.
// ---------------------------------------------------------------------------
__global__ __launch_bounds__(256) void crf_emit_gemm(
    const int* __restrict__ sents, const float* __restrict__ emb,
    const float* __restrict__ W, const float* __restrict__ bias,
    float* __restrict__ emit)
{
    __shared__ float ldsWp[E_ * K_];   // 32 KB, [p][n][q] with p=k/2, q=k&1
    for (int i = threadIdx.x; i < E_ * K_; i += 256) {
        const int k = i >> 5;          // 0..255
        const int n = i & 31;          // 0..31
        ldsWp[(k >> 1) * 64 + n * 2 + (k & 1)] = W[i];
    }
    __syncthreads();

    const int lane  = threadIdx.x & 31;
    const int wave  = threadIdx.x >> 5;
    const int group = blockIdx.x * 8 + wave;       // 16-row group of B*T rows
    const int rowbase = group * 16;

    // A-matrix layout (16x4 f32): lanes 0-15 hold K=0,1; lanes 16-31 hold K=2,3
    const int m    = lane & 15;                    // M index (row within group)
    const int koff = (lane >> 4) * 2;              // K sub-offset for this lane half
    const int n0   = lane & 15;                    // N index within a 16-wide tile

    const int sidx = sents[rowbase + m];           // gathered vocab row (PAD=0 is valid)
    const float* aptr = emb + (size_t)sidx * E_ + koff;

    // per-lane LDS base: pair index offset (koff/2)*64 dwords + n slot
    const float* bptr = &ldsWp[(koff >> 1) * 64 + n0 * 2];

    v8f acc0 = {};   // N = 0..15
    v8f acc1 = {};   // N = 16..31

    #pragma unroll
    for (int kk = 0; kk < E_; kk += 4) {
        v2f a  = *(const v2f*)(aptr + kk);             // global_load_b64
        v2f b0 = *(const v2f*)(bptr + kk * 32);        // ds_load_b64, imm offset kk*128B
        v2f b1 = *(const v2f*)(bptr + kk * 32 + 32);   // ds_load_b64, +128B (n+16 slot)

        acc0 = __builtin_amdgcn_wmma_f32_16x16x4_f32(false, a, false, b0,
                                                     (short)0, acc0, false, false);
        acc1 = __builtin_amdgcn_wmma_f32_16x16x4_f32(false, a, false, b1,
                                                     (short)0, acc1, false, false);
    }

    // D layout: VGPR r -> M = r (lanes 0-15) or r+8 (lanes 16-31); N = lane&15
    const int mhi = (lane >> 4) * 8;
    #pragma unroll
    for (int r = 0; r < 8; ++r) {
        const int M = r + mhi;
        float* dst = emit + (size_t)(rowbase + M) * K_;
        dst[n0]      = acc0[r] + bias[n0];
        dst[n0 + 16] = acc1[r] + bias[n0 + 16];
    }
}

// ---------------------------------------------------------------------------
// Kernel 2: total_score[b] = sum_t mask * (emit[b,t,tag_t] + trans[tag_{t-1},tag_t])
// ---------------------------------------------------------------------------
__global__ __launch_bounds__(256) void crf_score(
    const int* __restrict__ sents, const int* __restrict__ tags,
    const float* __restrict__ trans, const float* __restrict__ emit,
    float* __restrict__ total)
{
    const int b   = blockIdx.x;
    const int tid = threadIdx.x;
    const int base = b * T_;

    float s = 0.f;
    for (int t = tid; t < T_; t += 256) {
        if (sents[base + t] != 0) {                 // mask (PAD == 0)
            const int tg = tags[base + t];
            float v = emit[(size_t)(base + t) * K_ + tg];
            if (t > 0) v += trans[tags[base + t - 1] * K_ + tg];
            s += v;
        }
    }

    // wave reduce, then cross-wave reduce via LDS
    #pragma unroll
    for (int off = 16; off > 0; off >>= 1) s += __shfl_xor(s, off, 32);

    __shared__ float red[8];
    if ((tid & 31) == 0) red[tid >> 5] = s;
    __syncthreads();
    if (tid == 0) {
        float v = 0.f;
        #pragma unroll
        for (int i = 0; i < 8; ++i) v += red[i];
        total[b] = v;
    }
}

// ---------------------------------------------------------------------------
// Kernel 3: forward algorithm (logsumexp scan) — one wave per batch, lane = tag k.
// d[j] broadcast via v_readlane; transition column held in 32 VGPRs per lane.
// Mask is a prefix, so scan early-exits at the sequence length.
// ---------------------------------------------------------------------------
__global__ __launch_bounds__(32) void crf_forward(
    const int* __restrict__ sents, const float* __restrict__ trans,
    const float* __restrict__ emit, const float* __restrict__ total,
    float* __restrict__ loss)
{
    const int b = blockIdx.x;
    const int k = threadIdx.x;            // 0..31 == tag index

    float c[32];
    #pragma unroll
    for (int j = 0; j < 32; ++j) c[j] = trans[j * K_ + k];   // column trans[:,k]

    const size_t base = (size_t)b * T_;
    float d = emit[base * K_ + k];        // d0 = emit[:,0]

    for (int t = 1; t < T_; ++t) {
        if (sents[base + t] == 0) break;  // prefix mask: all later t are padded too

        const float e = emit[(base + t) * K_ + k];

        float x[32];
        float mx = -3.402823466e+38f;
        #pragma unroll
        for (int j = 0; j < 32; ++j) {
            const float dj = __int_as_float(
                __builtin_amdgcn_readlane(__float_as_int(d), j));
            x[j] = dj + c[j];
            mx = fmaxf(mx, x[j]);
        }
        float s = 0.f;
        #pragma unroll
        for (int j = 0; j < 32; ++j) s += __expf(x[j] - mx);

        d = e + mx + __logf(s);
    }

    // log_z = logsumexp over lanes of d
    float mx = d;
    #pragma unroll
    for (int off = 16; off > 0; off >>= 1) mx = fmaxf(mx, __shfl_xor(mx, off, 32));
    float s = __expf(d - mx);
    #pragma unroll
    for (int off = 16; off > 0; off >>= 1) s += __shfl_xor(s, off, 32);

    if (k == 0) loss[b] = mx + __logf(s) - total[b];
}

// ---------------------------------------------------------------------------
extern "C" void kernel_launch(void* const* d_in, const int* in_sizes, int n_in,
                              void* d_out, int out_size, void* d_ws, size_t ws_size,
                              hipStream_t stream)
{
    const int*   sents = (const int*)d_in[0];
    const int*   tags  = (const int*)d_in[1];
    const float* emb   = (const float*)d_in[2];
    const float* W     = (const float*)d_in[3];
    const float* bias  = (const float*)d_in[4];
    const float* trans = (const float*)d_in[5];
    float*       out   = (float*)d_out;

    float* emit  = (float*)d_ws;                       // B*T*K floats = 8 MB
    float* total = emit + (size_t)B_ * T_ * K_;        // B floats

    // 65536 rows / 16 rows-per-wave / 8 waves-per-block = 512 blocks
    crf_emit_gemm<<<(B_ * T_ / 16) / 8, 256, 0, stream>>>(sents, emb, W, bias, emit);
    crf_score<<<B_, 256, 0, stream>>>(sents, tags, trans, emit, total);
    crf_forward<<<B_, 32, 0, stream>>>(sents, trans, emit, total, out);
}